// Enhence_65730179498739
// MI455X (gfx1250) — compile-verified
//
#include <hip/hip_runtime.h>
#include <math.h>

#define BB   16
#define CC   512
#define HWD  4096
#define EPSF 1e-8f
#define NTOP 12

// workspace offsets (in floats)
#define WS_PREDF 0
#define WS_PREDB 65536
#define WS_NA    131072
#define WS_COSF  196608
#define WS_COSB  262144
#define WS_ACCF  327680
#define WS_ACCB  335872
#define WS_QFP   344064
#define WS_QBP   352256
#define WS_NQ    360448   // [fg:16][bg:16]
#define WS_CNT   360480   // 32 ints: [fg:16][bg:16]
#define WS_PMINF 360512   // 512 partials each
#define WS_PMAXF 361024
#define WS_PMINB 361536
#define WS_PMAXB 362048
#define WS_FIN   362560   // minF, invF, minB, invB

typedef float v2f __attribute__((ext_vector_type(2)));
typedef float v8f __attribute__((ext_vector_type(8)));

#if defined(__AMDGCN__) && __has_builtin(__builtin_amdgcn_wmma_f32_16x16x4_f32)
#define HAVE_WMMA 1
#endif

__device__ __forceinline__ v8f wmma_f32_16x16x4(v2f a, v2f b, v8f c) {
#ifdef HAVE_WMMA
  // 8 args: (neg_a, A, neg_b, B, c_mod, C, reuse_a, reuse_b)
  return __builtin_amdgcn_wmma_f32_16x16x4_f32(false, a, false, b, (short)0, c,
                                               false, false);
#else
  (void)a; (void)b;
  return c;  // host-compile / non-gfx1250 stub (never executed on device)
#endif
}

// ---------------------------------------------------------------------------
// K1: per-pixel cosine vs supp_fp/supp_bp via WMMA, softmax -> pred, store ||feat||
// grid: B * (HW/128) = 512 blocks, 256 threads = 8 waves, 16 pixels per wave
// A (16x4 f32): lane m (<16) holds A[m, 2h], A[m, 2h+1] for half h = lane>>4.
// B (4x16): lane m holds B[2h, m], B[2h+1, m]; only cols 0 (fg) / 1 (bg) nonzero.
// ---------------------------------------------------------------------------
__global__ __launch_bounds__(256) void k1_sim_pred(
    const float* __restrict__ feat, const float* __restrict__ sfp,
    const float* __restrict__ sbp, float* __restrict__ ws) {
  const int b    = blockIdx.x >> 5;
  const int tile = (blockIdx.x & 31) << 7;
  const int wave = threadIdx.x >> 5;
  const int lane = threadIdx.x & 31;
  const int half = lane >> 4;
  const int m    = lane & 15;
  const int pix  = tile + (wave << 4) + m;
  const float* fb = feat + (size_t)b * CC * HWD + pix + (size_t)(half << 1) * HWD;
  // branch-free B fragment: lane-selected pointer + multiplicative column mask
  const float* psel = (m == 1) ? (sbp + b * CC) : (sfp + b * CC);
  const float  cmask = (m < 2) ? 1.f : 0.f;

  v8f acc = {0.f, 0.f, 0.f, 0.f, 0.f, 0.f, 0.f, 0.f};
  float nrm = 0.f, nprot = 0.f;
#pragma unroll 8
  for (int k0 = 0; k0 < CC; k0 += 4) {
    v2f a;
    a.x = fb[(size_t)k0 * HWD];
    a.y = fb[(size_t)(k0 + 1) * HWD];
    const v2f praw = *(const v2f*)(psel + k0 + (half << 1));
    v2f bv;
    bv.x = praw.x * cmask;
    bv.y = praw.y * cmask;
    acc = wmma_f32_16x16x4(a, bv, acc);
    nrm   += a.x * a.x + a.y * a.y;
    nprot += bv.x * bv.x + bv.y * bv.y;
  }
  // combine half-lane partials: lane m and m+16 cover complementary channels
  const float nrmF = nrm + __shfl_xor(nrm, 16, 32);
  const float npF  = nprot + __shfl_xor(nprot, 16, 32);
  const float nsf  = sqrtf(__shfl(npF, 0, 32));  // ||supp_fp[b]||
  const float nsb  = sqrtf(__shfl(npF, 1, 32));  // ||supp_bp[b]||

  // D layout: lane0 holds col0 (fg dot) rows 0..7, lane16 rows 8..15; lane1/17 col1 (bg)
  __shared__ float ldsFG[8][16];
  __shared__ float ldsBG[8][16];
  if (lane == 0) {
#pragma unroll
    for (int j = 0; j < 8; ++j) ldsFG[wave][j] = acc[j];
  } else if (lane == 16) {
#pragma unroll
    for (int j = 0; j < 8; ++j) ldsFG[wave][8 + j] = acc[j];
  } else if (lane == 1) {
#pragma unroll
    for (int j = 0; j < 8; ++j) ldsBG[wave][j] = acc[j];
  } else if (lane == 17) {
#pragma unroll
    for (int j = 0; j < 8; ++j) ldsBG[wave][8 + j] = acc[j];
  }
  __syncthreads();
  if (lane < 16) {
    const float dfg = ldsFG[wave][m];
    const float dbg = ldsBG[wave][m];
    const float na  = sqrtf(nrmF);
    const float den = fmaxf(na, EPSF);
    const float sf  = dfg / (den * fmaxf(nsf, EPSF));
    const float sb  = dbg / (den * fmaxf(nsb, EPSF));
    // softmax over 2 channels of 10*[sim_bg, sim_fg]
    const float pfg = 1.f / (1.f + expf(10.f * (sb - sf)));
    const float pbg = 1.f / (1.f + expf(10.f * (sf - sb)));
    const int g = b * HWD + pix;
    ws[WS_PREDF + g] = pfg;
    ws[WS_PREDB + g] = pbg;
    ws[WS_NA + g]    = na;
  }
}

// ---------------------------------------------------------------------------
// K2: masked prototype sums. grid: B*C blocks, one (b,c) per block.
// ---------------------------------------------------------------------------
__global__ __launch_bounds__(256) void k2_proto_acc(
    const float* __restrict__ feat, const float* __restrict__ tau,
    float* __restrict__ ws, int* __restrict__ cnt) {
  const int b = blockIdx.x >> 9;
  const int c = blockIdx.x & (CC - 1);
  const int t = threadIdx.x;
  const float thF = 1.f / (1.f + expf(-tau[0]));
  const float thB = 1.f - thF;
  const float4* row = (const float4*)(feat + ((size_t)b * CC + c) * HWD);
  const float4* prF = (const float4*)(ws + WS_PREDF + b * HWD);
  const float4* prB = (const float4*)(ws + WS_PREDB + b * HWD);
  float sF = 0.f, sB = 0.f;
  int cF = 0, cB = 0;
  for (int i = t; i < (HWD >> 2); i += 256) {
    const float4 v = row[i], f = prF[i], g = prB[i];
    if (f.x > thF) { sF += v.x; ++cF; }  if (g.x > thB) { sB += v.x; ++cB; }
    if (f.y > thF) { sF += v.y; ++cF; }  if (g.y > thB) { sB += v.y; ++cB; }
    if (f.z > thF) { sF += v.z; ++cF; }  if (g.z > thB) { sB += v.z; ++cB; }
    if (f.w > thF) { sF += v.w; ++cF; }  if (g.w > thB) { sB += v.w; ++cB; }
  }
  __shared__ float rF[256], rB[256];
  __shared__ int   iF[256], iB[256];
  rF[t] = sF; rB[t] = sB; iF[t] = cF; iB[t] = cB;
  __syncthreads();
  for (int s = 128; s > 0; s >>= 1) {
    if (t < s) {
      rF[t] += rF[t + s]; rB[t] += rB[t + s];
      iF[t] += iF[t + s]; iB[t] += iB[t + s];
    }
    __syncthreads();
  }
  if (t == 0) {
    ws[WS_ACCF + b * CC + c] = rF[0];
    ws[WS_ACCB + b * CC + c] = rB[0];
    if (c == 0) { cnt[b] = iF[0]; cnt[BB + b] = iB[0]; }
  }
}

// ---------------------------------------------------------------------------
// K3: top-12 fallback + final prototypes + prototype norms. grid: B blocks.
// ---------------------------------------------------------------------------
__global__ __launch_bounds__(256) void k3_topk_proto(
    const float* __restrict__ feat, float* __restrict__ ws,
    const int* __restrict__ cnt, float* __restrict__ out_qfp,
    float* __restrict__ out_qbp) {
  const int b = blockIdx.x;
  const int t = threadIdx.x;
  __shared__ float vals[HWD];   // 16 KB
  __shared__ float red[256];
  __shared__ int   redi[256];
  __shared__ int   idxs[NTOP];
  for (int pass = 0; pass < 2; ++pass) {
    const float* pred = ws + (pass == 0 ? WS_PREDF : WS_PREDB) + b * HWD;
    for (int p = t; p < HWD; p += 256) vals[p] = pred[p];
    __syncthreads();
    for (int k = 0; k < NTOP; ++k) {
      float best = -1e30f; int bi = 0;
      for (int p = t; p < HWD; p += 256) {
        const float v = vals[p];
        if (v > best) { best = v; bi = p; }
      }
      red[t] = best; redi[t] = bi;
      __syncthreads();
      for (int s = 128; s > 0; s >>= 1) {
        if (t < s) {
          if (red[t + s] > red[t] ||
              (red[t + s] == red[t] && redi[t + s] < redi[t])) {
            red[t] = red[t + s]; redi[t] = redi[t + s];
          }
        }
        __syncthreads();
      }
      if (t == 0) { idxs[k] = redi[0]; vals[redi[0]] = -1e30f; }
      __syncthreads();
    }
    const int cN = cnt[pass * BB + b];
    const float invc = 1.f / (float)(cN > 0 ? cN : 1);
    const float* accp = ws + (pass == 0 ? WS_ACCF : WS_ACCB) + b * CC;
    float* qws  = ws + (pass == 0 ? WS_QFP : WS_QBP) + b * CC;
    float* qout = (pass == 0 ? out_qfp : out_qbp) + b * CC;
    float n2 = 0.f;
    for (int c = t; c < CC; c += 256) {
      float tm = 0.f;
      for (int k = 0; k < NTOP; ++k)
        tm += feat[((size_t)b * CC + c) * HWD + idxs[k]];
      tm *= (1.f / NTOP);
      const float q = (cN > 0) ? accp[c] * invc : tm;
      qws[c] = q; qout[c] = q;
      n2 += q * q;
    }
    red[t] = n2;
    __syncthreads();
    for (int s = 128; s > 0; s >>= 1) {
      if (t < s) red[t] += red[t + s];
      __syncthreads();
    }
    if (t == 0) ws[WS_NQ + pass * BB + b] = sqrtf(red[0]);
    __syncthreads();
  }
}

// ---------------------------------------------------------------------------
// K4: cosine vs query prototypes via WMMA + block min/max partials.
// ---------------------------------------------------------------------------
__global__ __launch_bounds__(256) void k4_cos(
    const float* __restrict__ feat, float* __restrict__ ws) {
  const int b    = blockIdx.x >> 5;
  const int tile = (blockIdx.x & 31) << 7;
  const int wave = threadIdx.x >> 5;
  const int lane = threadIdx.x & 31;
  const int half = lane >> 4;
  const int m    = lane & 15;
  const int pix  = tile + (wave << 4) + m;
  const float* fb = feat + (size_t)b * CC * HWD + pix + (size_t)(half << 1) * HWD;
  const float* psel = ws + (m == 1 ? WS_QBP : WS_QFP) + b * CC;
  const float  cmask = (m < 2) ? 1.f : 0.f;

  v8f acc = {0.f, 0.f, 0.f, 0.f, 0.f, 0.f, 0.f, 0.f};
#pragma unroll 8
  for (int k0 = 0; k0 < CC; k0 += 4) {
    v2f a;
    a.x = fb[(size_t)k0 * HWD];
    a.y = fb[(size_t)(k0 + 1) * HWD];
    const v2f praw = *(const v2f*)(psel + k0 + (half << 1));
    v2f bv;
    bv.x = praw.x * cmask;
    bv.y = praw.y * cmask;
    acc = wmma_f32_16x16x4(a, bv, acc);
  }
  const float nqf = fmaxf(ws[WS_NQ + b], EPSF);
  const float nqb = fmaxf(ws[WS_NQ + BB + b], EPSF);

  float lminF = 1e30f, lmaxF = -1e30f, lminB = 1e30f, lmaxB = -1e30f;
  const int rowbase = b * HWD + tile + (wave << 4) + ((lane & 16) ? 8 : 0);
  if (m == 0) {  // lanes 0/16: column 0 = fg numerators, rows 0..7 / 8..15
#pragma unroll
    for (int j = 0; j < 8; ++j) {
      const float na = fmaxf(ws[WS_NA + rowbase + j], EPSF);
      const float cf = acc[j] / (na * nqf);
      ws[WS_COSF + rowbase + j] = cf;
      lminF = fminf(lminF, cf); lmaxF = fmaxf(lmaxF, cf);
    }
  } else if (m == 1) {  // lanes 1/17: column 1 = bg numerators
#pragma unroll
    for (int j = 0; j < 8; ++j) {
      const float na = fmaxf(ws[WS_NA + rowbase + j], EPSF);
      const float cb = acc[j] / (na * nqb);
      ws[WS_COSB + rowbase + j] = cb;
      lminB = fminf(lminB, cb); lmaxB = fmaxf(lmaxB, cb);
    }
  }
  lminF = fminf(lminF, __shfl_xor(lminF, 16, 32));
  lmaxF = fmaxf(lmaxF, __shfl_xor(lmaxF, 16, 32));
  lminB = fminf(lminB, __shfl_xor(lminB, 16, 32));
  lmaxB = fmaxf(lmaxB, __shfl_xor(lmaxB, 16, 32));

  __shared__ float sMinF[8], sMaxF[8], sMinB[8], sMaxB[8];
  if (lane == 0) { sMinF[wave] = lminF; sMaxF[wave] = lmaxF; }
  if (lane == 1) { sMinB[wave] = lminB; sMaxB[wave] = lmaxB; }
  __syncthreads();
  if (threadIdx.x == 0) {
    float mnF = 1e30f, mxF = -1e30f, mnB = 1e30f, mxB = -1e30f;
    for (int w = 0; w < 8; ++w) {
      mnF = fminf(mnF, sMinF[w]); mxF = fmaxf(mxF, sMaxF[w]);
      mnB = fminf(mnB, sMinB[w]); mxB = fmaxf(mxB, sMaxB[w]);
    }
    ws[WS_PMINF + blockIdx.x] = mnF;
    ws[WS_PMAXF + blockIdx.x] = mxF;
    ws[WS_PMINB + blockIdx.x] = mnB;
    ws[WS_PMAXB + blockIdx.x] = mxB;
  }
}

// ---------------------------------------------------------------------------
// K5: reduce 512 block partials -> global min + inverse range.
// ---------------------------------------------------------------------------
__global__ __launch_bounds__(256) void k5_minmax(float* __restrict__ ws) {
  const int t = threadIdx.x;
  float mnF = 1e30f, mxF = -1e30f, mnB = 1e30f, mxB = -1e30f;
  for (int i = t; i < 512; i += 256) {
    mnF = fminf(mnF, ws[WS_PMINF + i]); mxF = fmaxf(mxF, ws[WS_PMAXF + i]);
    mnB = fminf(mnB, ws[WS_PMINB + i]); mxB = fmaxf(mxB, ws[WS_PMAXB + i]);
  }
  __shared__ float rA[256], rB[256], rC[256], rD[256];
  rA[t] = mnF; rB[t] = mxF; rC[t] = mnB; rD[t] = mxB;
  __syncthreads();
  for (int s = 128; s > 0; s >>= 1) {
    if (t < s) {
      rA[t] = fminf(rA[t], rA[t + s]); rB[t] = fmaxf(rB[t], rB[t + s]);
      rC[t] = fminf(rC[t], rC[t + s]); rD[t] = fmaxf(rD[t], rD[t + s]);
    }
    __syncthreads();
  }
  if (t == 0) {
    ws[WS_FIN + 0] = rA[0];
    ws[WS_FIN + 1] = 1.f / (rB[0] - rA[0]);
    ws[WS_FIN + 2] = rC[0];
    ws[WS_FIN + 3] = 1.f / (rD[0] - rC[0]);
  }
}

// ---------------------------------------------------------------------------
// K6: out = feat * (act + 1 - deact), float4 vectorized.
// ---------------------------------------------------------------------------
__global__ __launch_bounds__(256) void k6_blend(
    const float* __restrict__ feat, const float* __restrict__ ws,
    float* __restrict__ out) {
  const size_t base = ((size_t)blockIdx.x * 256 + threadIdx.x) * 4;
  const int p = (int)(base & (size_t)(HWD - 1));
  const int b = (int)(base >> 21);  // / (C*HW) = 2^21
  const int g = b * HWD + p;
  const float mnF = ws[WS_FIN + 0], ivF = ws[WS_FIN + 1];
  const float mnB = ws[WS_FIN + 2], ivB = ws[WS_FIN + 3];
  const float4 v  = *(const float4*)(feat + base);
  const float4 cf = *(const float4*)(ws + WS_COSF + g);
  const float4 cb = *(const float4*)(ws + WS_COSB + g);
  float4 o;
  o.x = v.x * ((cf.x - mnF) * ivF + 1.f - (cb.x - mnB) * ivB);
  o.y = v.y * ((cf.y - mnF) * ivF + 1.f - (cb.y - mnB) * ivB);
  o.z = v.z * ((cf.z - mnF) * ivF + 1.f - (cb.z - mnB) * ivB);
  o.w = v.w * ((cf.w - mnF) * ivF + 1.f - (cb.w - mnB) * ivB);
  *(float4*)(out + base) = o;
}

extern "C" void kernel_launch(void* const* d_in, const int* in_sizes, int n_in,
                              void* d_out, int out_size, void* d_ws,
                              size_t ws_size, hipStream_t stream) {
  (void)in_sizes; (void)n_in; (void)out_size; (void)ws_size;
  const float* sfp  = (const float*)d_in[0];  // (B,C,1,1)
  const float* sbp  = (const float*)d_in[1];  // (B,C,1,1)
  const float* feat = (const float*)d_in[2];  // (B,C,H,W)
  const float* tau  = (const float*)d_in[3];  // scalar
  float* out = (float*)d_out;
  float* ws  = (float*)d_ws;
  float* out_qfp = out + (size_t)BB * CC * HWD;
  float* out_qbp = out_qfp + BB * CC;
  int* cnt = (int*)(ws + WS_CNT);

  k1_sim_pred<<<BB * (HWD / 128), 256, 0, stream>>>(feat, sfp, sbp, ws);
  k2_proto_acc<<<BB * CC, 256, 0, stream>>>(feat, tau, ws, cnt);
  k3_topk_proto<<<BB, 256, 0, stream>>>(feat, ws, cnt, out_qfp, out_qbp);
  k4_cos<<<BB * (HWD / 128), 256, 0, stream>>>(feat, ws);
  k5_minmax<<<1, 256, 0, stream>>>(ws);
  k6_blend<<<(BB * CC * HWD) / 1024, 256, 0, stream>>>(feat, ws, out);
}